// GraphAttentionNetwork_16509854286164
// MI455X (gfx1250) — compile-verified
//
#include <hip/hip_runtime.h>
#include <math.h>

#define N_NODES   50000
#define DIMC      64
#define H_HEADS   8
#define E_EDGES   800000
#define M_EDGES   (E_EDGES + N_NODES)   // edges + self loops
#define HD        (H_HEADS * DIMC)      // 512
#define NEG_SLOPE 0.2f

typedef __attribute__((ext_vector_type(2))) float v2f;
typedef __attribute__((ext_vector_type(8))) float v8f;

// ---------- order-preserving float<->uint encoding for atomicMax ----------
__device__ __forceinline__ unsigned f_enc(float f) {
  unsigned u = __float_as_uint(f);
  return (u & 0x80000000u) ? ~u : (u | 0x80000000u);
}
__device__ __forceinline__ float f_dec(unsigned u) {
  return __uint_as_float((u & 0x80000000u) ? (u & 0x7FFFFFFFu) : ~u);
}
__device__ __forceinline__ float lrelu(float v) {
  return (v > 0.f) ? v : NEG_SLOPE * v;
}

// ---------- 1) projection GEMM: h[N,512] = x[N,64] @ W[64,512] (fp32 WMMA) ----------
// One wave computes a 16x64 strip (4 adjacent 16x16 tiles) -> A fragment reused 4x.
__global__ __launch_bounds__(256) void gat_gemm_wmma(const float* __restrict__ x,
                                                     const float* __restrict__ W,
                                                     float* __restrict__ h) {
  const int wave = blockIdx.x * 8 + (threadIdx.x >> 5);  // 0..24999
  const int lane = threadIdx.x & 31;
  const int hl   = lane >> 4;     // half-wave: 0 or 1
  const int l    = lane & 15;
  const int tm   = wave >> 3;     // row tile 0..3124
  const int tng  = wave & 7;      // group of 4 col tiles (64 cols)

  const float* xrow  = x + (size_t)(tm * 16 + l) * DIMC;  // A: row M=l
  const float* wbase = W + tng * 64 + l;                  // B: 4 col tiles

  v8f c0 = {}, c1 = {}, c2 = {}, c3 = {};
#pragma unroll
  for (int k = 0; k < DIMC; k += 4) {
    // A 16x4 fp32: VGPR0 = K(0|2), VGPR1 = K(1|3) per half-wave -> contiguous pair
    v2f a;
    const float* ap = xrow + k + 2 * hl;
    a.x = ap[0];
    a.y = ap[1];
    // B 4x16 fp32: VGPR i holds row K = i + 2*half, N striped across lanes
    const float* bp = wbase + (size_t)(k + 2 * hl) * HD;
    v2f b0, b1, b2, b3;
    b0.x = bp[0];  b0.y = bp[HD];
    b1.x = bp[16]; b1.y = bp[HD + 16];
    b2.x = bp[32]; b2.y = bp[HD + 32];
    b3.x = bp[48]; b3.y = bp[HD + 48];
    c0 = __builtin_amdgcn_wmma_f32_16x16x4_f32(false, a, false, b0, (short)0, c0, false, false);
    c1 = __builtin_amdgcn_wmma_f32_16x16x4_f32(false, a, false, b1, (short)0, c1, false, false);
    c2 = __builtin_amdgcn_wmma_f32_16x16x4_f32(false, a, false, b2, (short)0, c2, false, false);
    c3 = __builtin_amdgcn_wmma_f32_16x16x4_f32(false, a, false, b3, (short)0, c3, false, false);
  }
  // C/D 16x16 f32: VGPR i -> row (i + 8*half), col = l
  float* hp = h + (size_t)(tm * 16) * HD + tng * 64 + l;
#pragma unroll
  for (int i = 0; i < 8; ++i) {
    float* row = hp + (size_t)(i + 8 * hl) * HD;
    row[0]  = c0[i];
    row[16] = c1[i];
    row[32] = c2[i];
    row[48] = c3[i];
  }
}

// ---------- 2) attention dots a_src/a_dst[N*H] + init segmax/denom ----------
__global__ void gat_att(const float* __restrict__ h, const float* __restrict__ att_src,
                        const float* __restrict__ att_dst, float* __restrict__ a_src,
                        float* __restrict__ a_dst, unsigned* __restrict__ seg_max,
                        float* __restrict__ denom) {
  int idx = blockIdx.x * blockDim.x + threadIdx.x;   // idx = n*H + head
  if (idx >= N_NODES * H_HEADS) return;
  int hh = idx & (H_HEADS - 1);
  const float4* hp = (const float4*)(h + (size_t)idx * DIMC);  // h[n,head,:]
  const float4* as = (const float4*)(att_src + hh * DIMC);
  const float4* ad = (const float4*)(att_dst + hh * DIMC);
  float s = 0.f, d = 0.f;
#pragma unroll
  for (int c = 0; c < DIMC / 4; ++c) {
    float4 v = hp[c], p = as[c], q = ad[c];
    s += v.x * p.x + v.y * p.y + v.z * p.z + v.w * p.w;
    d += v.x * q.x + v.y * q.y + v.z * q.z + v.w * q.w;
  }
  a_src[idx] = s;
  a_dst[idx] = d;
  seg_max[idx] = 0u;   // encoded "below any real float"
  denom[idx] = 0.f;
}

// ---------- 3) out = bias (broadcast) ----------
__global__ void gat_init_out(const float* __restrict__ bias, float* __restrict__ out) {
  int idx = blockIdx.x * blockDim.x + threadIdx.x;
  if (idx >= N_NODES * DIMC) return;
  out[idx] = bias[idx & (DIMC - 1)];
}

// ---------- 4) edge logits (leaky relu) + segment max; one thread per edge ----------
__global__ void gat_logits(const int* __restrict__ ei, const float* __restrict__ a_src,
                           const float* __restrict__ a_dst, float* __restrict__ ebuf,
                           unsigned* __restrict__ seg_max) {
  int e = blockIdx.x * blockDim.x + threadIdx.x;
  if (e >= M_EDGES) return;
  int s = (e < E_EDGES) ? ei[e] : (e - E_EDGES);
  int d = (e < E_EDGES) ? ei[E_EDGES + e] : (e - E_EDGES);
  const float4* asv = (const float4*)(a_src + s * H_HEADS);
  const float4* adv = (const float4*)(a_dst + d * H_HEADS);
  float4* eb = (float4*)(ebuf + (size_t)e * H_HEADS);
  unsigned* sm = seg_max + d * H_HEADS;
#pragma unroll
  for (int j = 0; j < 2; ++j) {
    float4 v = asv[j], w = adv[j];
    float4 r;
    r.x = lrelu(v.x + w.x);
    r.y = lrelu(v.y + w.y);
    r.z = lrelu(v.z + w.z);
    r.w = lrelu(v.w + w.w);
    eb[j] = r;
    atomicMax(&sm[j * 4 + 0], f_enc(r.x));
    atomicMax(&sm[j * 4 + 1], f_enc(r.y));
    atomicMax(&sm[j * 4 + 2], f_enc(r.z));
    atomicMax(&sm[j * 4 + 3], f_enc(r.w));
  }
}

// ---------- 5) exp(logit - max), denom scatter-add; one thread per edge ----------
__global__ void gat_expsum(const int* __restrict__ ei, float* __restrict__ ebuf,
                           const unsigned* __restrict__ seg_max,
                           float* __restrict__ denom) {
  int e = blockIdx.x * blockDim.x + threadIdx.x;
  if (e >= M_EDGES) return;
  int d = (e < E_EDGES) ? ei[E_EDGES + e] : (e - E_EDGES);
  float4* eb = (float4*)(ebuf + (size_t)e * H_HEADS);
  const unsigned* sm = seg_max + d * H_HEADS;
  float* dn = denom + d * H_HEADS;
#pragma unroll
  for (int j = 0; j < 2; ++j) {
    float4 lv = eb[j];
    float m0 = f_dec(sm[j * 4 + 0]);
    float m1 = f_dec(sm[j * 4 + 1]);
    float m2 = f_dec(sm[j * 4 + 2]);
    float m3 = f_dec(sm[j * 4 + 3]);
    if (!__builtin_isfinite(m0)) m0 = 0.f;
    if (!__builtin_isfinite(m1)) m1 = 0.f;
    if (!__builtin_isfinite(m2)) m2 = 0.f;
    if (!__builtin_isfinite(m3)) m3 = 0.f;
    float4 ev;
    ev.x = __expf(lv.x - m0);
    ev.y = __expf(lv.y - m1);
    ev.z = __expf(lv.z - m2);
    ev.w = __expf(lv.w - m3);
    eb[j] = ev;
    atomicAdd(&dn[j * 4 + 0], ev.x);
    atomicAdd(&dn[j * 4 + 1], ev.y);
    atomicAdd(&dn[j * 4 + 2], ev.z);
    atomicAdd(&dn[j * 4 + 3], ev.w);
  }
}

// ---------- 6) weighted message scatter: one wave32 per edge ----------
// Each lane owns 2 output channels, accumulates all 8 heads in registers,
// then issues only 2 atomics -> 64 atomics/edge instead of 512.
__global__ __launch_bounds__(256) void gat_scatter(const int* __restrict__ ei,
                                                   const float* __restrict__ h,
                                                   const float* __restrict__ ebuf,
                                                   const float* __restrict__ denom,
                                                   float* __restrict__ out) {
  int e = blockIdx.x * 8 + (threadIdx.x >> 5);
  if (e >= M_EDGES) return;
  int lane = threadIdx.x & 31;
  int s = (e < E_EDGES) ? ei[e] : (e - E_EDGES);
  int d = (e < E_EDGES) ? ei[E_EDGES + e] : (e - E_EDGES);
  // lanes 0..7 compute per-head alpha (folded with 1/H)
  float alpha = 0.f;
  if (lane < H_HEADS)
    alpha = ebuf[(size_t)e * H_HEADS + lane] / denom[d * H_HEADS + lane] *
            (1.0f / H_HEADS);
  const float* hp = h + (size_t)s * HD;
  float acc0 = 0.f, acc1 = 0.f;
#pragma unroll
  for (int hh = 0; hh < H_HEADS; ++hh) {
    float a = __shfl(alpha, hh, 32);
    acc0 += a * hp[hh * DIMC + lane];
    acc1 += a * hp[hh * DIMC + lane + 32];
  }
  float* op = out + (size_t)d * DIMC;
  atomicAdd(&op[lane], acc0);
  atomicAdd(&op[lane + 32], acc1);
}

extern "C" void kernel_launch(void* const* d_in, const int* in_sizes, int n_in,
                              void* d_out, int out_size, void* d_ws, size_t ws_size,
                              hipStream_t stream) {
  const float* x       = (const float*)d_in[0];
  const int*   ei      = (const int*)d_in[1];
  const float* W       = (const float*)d_in[2];
  const float* att_src = (const float*)d_in[3];
  const float* att_dst = (const float*)d_in[4];
  const float* bias    = (const float*)d_in[5];
  float* out = (float*)d_out;

  char* ws = (char*)d_ws;
  size_t off = 0;
  auto alloc = [&](size_t bytes) -> void* {
    void* p = ws + off;
    off = (off + bytes + 255) & ~(size_t)255;
    return p;
  };
  float*    h      = (float*)alloc(sizeof(float) * (size_t)N_NODES * HD);       // 102.4 MB
  float*    a_src  = (float*)alloc(sizeof(float) * N_NODES * H_HEADS);          // 1.6 MB
  float*    a_dst  = (float*)alloc(sizeof(float) * N_NODES * H_HEADS);          // 1.6 MB
  unsigned* smax   = (unsigned*)alloc(sizeof(unsigned) * N_NODES * H_HEADS);    // 1.6 MB
  float*    denom  = (float*)alloc(sizeof(float) * N_NODES * H_HEADS);          // 1.6 MB
  float*    ebuf   = (float*)alloc(sizeof(float) * (size_t)M_EDGES * H_HEADS);  // 27.2 MB

  const int WAVES = (N_NODES / 16) * (HD / 64);   // 3125 * 8 = 25000 waves
  gat_gemm_wmma<<<WAVES / 8, 256, 0, stream>>>(x, W, h);
  gat_att<<<(N_NODES * H_HEADS + 255) / 256, 256, 0, stream>>>(h, att_src, att_dst,
                                                               a_src, a_dst, smax, denom);
  gat_init_out<<<(N_NODES * DIMC + 255) / 256, 256, 0, stream>>>(bias, out);
  gat_logits<<<(M_EDGES + 255) / 256, 256, 0, stream>>>(ei, a_src, a_dst, ebuf, smax);
  gat_expsum<<<(M_EDGES + 255) / 256, 256, 0, stream>>>(ei, ebuf, smax, denom);
  gat_scatter<<<(M_EDGES + 7) / 8, 256, 0, stream>>>(ei, h, ebuf, denom, out);
}